// FaceTokenVQ_5712306503811
// MI455X (gfx1250) — compile-verified
//
#include <hip/hip_runtime.h>
#include <stdint.h>

typedef __attribute__((ext_vector_type(2))) float v2f;
typedef __attribute__((ext_vector_type(8))) float v8f;
typedef __attribute__((ext_vector_type(4))) int   v4i;

#define HH      8
#define DD      16
#define KCODES  16384
#define BROWS   16384          // B*H = 2048*8
#define KDIM    192            // 16*7 + 16*5 feature dims
#define LDSPAD  196            // 192 + 4 pad -> distinct LDS bank groups per column
#define NTILES  (KCODES / 16)

// CDNA5 async global->LDS DMA path (ASYNCcnt-tracked), guarded: fall back to a
// register-double-buffered synchronous pipeline if the builtins are absent.
#if defined(__gfx1250__) && \
    __has_builtin(__builtin_amdgcn_global_load_async_to_lds_b128) && \
    __has_builtin(__builtin_amdgcn_s_wait_asynccnt)
#define USE_ASYNC_LDS 1
#else
#define USE_ASYNC_LDS 0
#endif

#if USE_ASYNC_LDS
typedef __attribute__((address_space(1))) v4i glob_v4i;
typedef __attribute__((address_space(3))) v4i  lds_v4i;

__device__ __forceinline__ void async_b128(const float* gsrc, float* ldst)
{
    __builtin_amdgcn_global_load_async_to_lds_b128(
        (glob_v4i*)(uintptr_t)gsrc,
        (lds_v4i*)(uintptr_t)ldst,
        0, 0);
}
#endif

// ---------------------------------------------------------------------------
// Kernel 1: e = equi_linear(w_eq, codebook); also emit per-code K-features.
//   e[c,o,k] = sum_i w[GRADES[k],o,i] * cb[c,i,k]
//   kfeat[c, o*7+j]        = e[c,o,IP_IDX[j]]
//   kfeat[c, 112 + o*5+j]  = bk-quadratic of normalized trivector part
// ---------------------------------------------------------------------------
__global__ __launch_bounds__(256) void build_e_kfeat(
    const float* __restrict__ cb, const float* __restrict__ w,
    float* __restrict__ e, float* __restrict__ kfeat)
{
    __shared__ float s_cb[16 * 256];   // 16 codes x (16x16)
    __shared__ float s_w[5 * 256];     // 5 grade matrices

    const int t  = threadIdx.x;
    const int c0 = blockIdx.x * 16;

    for (int j = t; j < 5 * 256; j += 256) s_w[j] = w[j];
    {
        const float4* src = (const float4*)(cb + (size_t)c0 * 256);
        float4*       dst = (float4*)s_cb;
        #pragma unroll
        for (int j = 0; j < 4; ++j) dst[t + j * 256] = src[t + j * 256];
    }
    __syncthreads();

    const int cl = t >> 4;
    const int o  = t & 15;
    const int GR[16] = {0,1,1,1,1,2,2,2,2,2,2,3,3,3,3,4};

    float er[16];
    #pragma unroll
    for (int k = 0; k < 16; ++k) {
        const float* wr = s_w  + GR[k] * 256 + o * 16;
        const float* cc = s_cb + cl * 256 + k;
        float acc = 0.f;
        #pragma unroll
        for (int i = 0; i < 16; ++i) acc += wr[i] * cc[i * 16];
        er[k] = acc;
    }

    const int c = c0 + cl;
    float* ep = e + ((size_t)c * 16 + o) * 16;
    #pragma unroll
    for (int k = 0; k < 16; ++k) ep[k] = er[k];

    const int IP[7] = {0,2,3,4,8,9,10};
    float* kp = kfeat + (size_t)c * KDIM;
    #pragma unroll
    for (int j = 0; j < 7; ++j) kp[o * 7 + j] = er[IP[j]];

    // bk basis: f0=-t3^2, f1=-(t0^2+t1^2+t2^2), f2..f4 = 2 ti t3
    const float r0 = er[11], r1 = er[12], r2 = er[13], r3 = er[14];
    const float s  = r3 / (r3 * r3 + 1e-3f);
    const float t0 = r0 * s, t1 = r1 * s, t2 = r2 * s, t3 = r3 * s;
    float* dp = kp + 112 + o * 5;
    dp[0] = -(t3 * t3);
    dp[1] = -(t0 * t0 + t1 * t1 + t2 * t2);
    dp[2] = 2.f * t0 * t3;
    dp[3] = 2.f * t1 * t3;
    dp[4] = 2.f * t2 * t3;
}

// ---------------------------------------------------------------------------
// Kernel 2: Q-features from x_flat.
//   bq basis: f0=t0^2+t1^2+t2^2, f1=t3^2, f2..f4 = ti t3
// ---------------------------------------------------------------------------
__global__ __launch_bounds__(256) void build_qfeat(
    const float* __restrict__ x, float* __restrict__ qfeat)
{
    const int gid = blockIdx.x * 256 + threadIdx.x;  // gid = r*16 + d
    const int r = gid >> 4;
    const int d = gid & 15;

    float xv[16];
    const float4* x4 = (const float4*)(x + (size_t)gid * 16);
    #pragma unroll
    for (int j = 0; j < 4; ++j) {
        float4 v = x4[j];
        xv[j * 4 + 0] = v.x; xv[j * 4 + 1] = v.y;
        xv[j * 4 + 2] = v.z; xv[j * 4 + 3] = v.w;
    }

    const int IP[7] = {0,2,3,4,8,9,10};
    float* qp = qfeat + (size_t)r * KDIM;
    #pragma unroll
    for (int j = 0; j < 7; ++j) qp[d * 7 + j] = xv[IP[j]];

    const float r0 = xv[11], r1 = xv[12], r2 = xv[13], r3 = xv[14];
    const float s  = r3 / (r3 * r3 + 1e-3f);
    const float t0 = r0 * s, t1 = r1 * s, t2 = r2 * s, t3 = r3 * s;
    float* dp = qp + 112 + d * 5;
    dp[0] = t0 * t0 + t1 * t1 + t2 * t2;
    dp[1] = t3 * t3;
    dp[2] = t0 * t3;
    dp[3] = t1 * t3;
    dp[4] = t2 * t3;
}

// ---------------------------------------------------------------------------
// Kernel 3: fused fp32 WMMA GEMM (Q @ K^T) + row argmax, double-buffered LDS.
// 8 waves/block; wave w owns rows [blk*128 + w*16, +16) and scans all 16384
// columns in 16-wide tiles. Tile i+1 is DMA'd (async global->LDS, ASYNCcnt)
// into the alternate buffer while 48 v_wmma_f32_16x16x4_f32 run on tile i.
// ---------------------------------------------------------------------------
__global__ __launch_bounds__(256) void dist_argmax(
    const float* __restrict__ qfeat, const float* __restrict__ kfeat,
    int* __restrict__ idx_out)
{
    __shared__ float s_k[2][16 * LDSPAD];

    const int t    = threadIdx.x;
    const int lane = t & 31;
    const int wave = t >> 5;
    const int row_base = blockIdx.x * 128 + wave * 16;

    const int m   = lane & 15;   // A row / C column selector
    const int kh  = lane >> 4;   // K half within each K=4 chunk
    const int col = lane & 15;

    // A fragments for this wave's 16 rows, resident for the whole N sweep.
    v2f a[48];
    {
        const float* qrow = qfeat + (size_t)(row_base + m) * KDIM + 2 * kh;
        #pragma unroll
        for (int q = 0; q < 48; ++q)
            a[q] = *(const v2f*)(qrow + 4 * q);
    }

    float vmax[8];
    int   vidx[8];
    #pragma unroll
    for (int s2 = 0; s2 < 8; ++s2) { vmax[s2] = -3.4e38f; vidx[s2] = 0; }

    // Staging map: thread t -> column lcol, 3 x b128 pieces at part+16*jj.
    const int lcol = t >> 4;
    const int part = t & 15;
    const float* gbase = kfeat + (size_t)lcol * KDIM + (size_t)part * 4;
    float* lbase0 = &s_k[0][lcol * LDSPAD + part * 4];
    float* lbase1 = &s_k[1][lcol * LDSPAD + part * 4];

    auto compute_tile = [&](int buf, int n0) {
        v8f c = {};
        const float* bp = &s_k[buf][col * LDSPAD + 2 * kh];
        #pragma unroll
        for (int q = 0; q < 48; ++q) {
            v2f b = *(const v2f*)(bp + 4 * q);
            c = __builtin_amdgcn_wmma_f32_16x16x4_f32(
                    false, a[q], false, b, (short)0, c, false, false);
        }
        #pragma unroll
        for (int s2 = 0; s2 < 8; ++s2) {
            if (c[s2] > vmax[s2]) { vmax[s2] = c[s2]; vidx[s2] = n0 + col; }
        }
    };

#if USE_ASYNC_LDS
    // -------- async DMA pipeline: 1 barrier + 1 asynccnt wait per tile -----
    {
        const float* gs = gbase;           // tile 0
        async_b128(gs,       lbase0);
        async_b128(gs + 64,  lbase0 + 64);
        async_b128(gs + 128, lbase0 + 128);
    }
    __builtin_amdgcn_s_wait_asynccnt(0);
    __syncthreads();

    for (int i = 0; i < NTILES; ++i) {
        if (i + 1 < NTILES) {
            const float* gs = gbase + (size_t)(i + 1) * 16 * KDIM;
            float* ld = ((i + 1) & 1) ? lbase1 : lbase0;
            async_b128(gs,       ld);
            async_b128(gs + 64,  ld + 64);
            async_b128(gs + 128, ld + 128);
        }
        compute_tile(i & 1, i * 16);
        __builtin_amdgcn_s_wait_asynccnt(0);
        __syncthreads();
    }
#else
    // -------- fallback: register-double-buffered synchronous pipeline ------
    {
        const float4* gs = (const float4*)gbase;   // tile 0
        float4 r0 = gs[0], r1 = gs[16], r2 = gs[32];
        float4* ld = (float4*)lbase0;
        ld[0] = r0; ld[16] = r1; ld[32] = r2;
    }
    __syncthreads();

    for (int i = 0; i < NTILES; ++i) {
        float4 r0, r1, r2;
        const bool more = (i + 1 < NTILES);
        if (more) {
            const float4* gs = (const float4*)(gbase + (size_t)(i + 1) * 16 * KDIM);
            r0 = gs[0]; r1 = gs[16]; r2 = gs[32];
        }
        compute_tile(i & 1, i * 16);
        if (more) {
            float4* ld = (float4*)(((i + 1) & 1) ? lbase1 : lbase0);
            ld[0] = r0; ld[16] = r1; ld[32] = r2;
        }
        __syncthreads();
    }
#endif

    // Butterfly reduce across the 16 lanes holding one row; ties -> lower idx
    // (matches jnp.argmax first-occurrence).
    #pragma unroll
    for (int mask = 1; mask < 16; mask <<= 1) {
        #pragma unroll
        for (int s2 = 0; s2 < 8; ++s2) {
            float om = __shfl_xor(vmax[s2], mask, 32);
            int   oi = __shfl_xor(vidx[s2], mask, 32);
            if (om > vmax[s2] || (om == vmax[s2] && oi < vidx[s2])) {
                vmax[s2] = om; vidx[s2] = oi;
            }
        }
    }
    if (col == 0) {
        #pragma unroll
        for (int s2 = 0; s2 < 8; ++s2)
            idx_out[row_base + kh * 8 + s2] = vidx[s2];
    }
}

// ---------------------------------------------------------------------------
// Kernel 4: gather e[i] and emit (e_sel, (e_sel - x) + x).
// ---------------------------------------------------------------------------
__global__ __launch_bounds__(256) void gather_out(
    const float* __restrict__ x, const float* __restrict__ e,
    const int* __restrict__ idx,
    float* __restrict__ out0, float* __restrict__ out1)
{
    const int gid = blockIdx.x * 256 + threadIdx.x;  // (b, hd, k) flat
    const int k  = gid & 15;
    const int hd = (gid >> 4) & 127;
    const int b  = gid >> 11;
    const int h  = hd >> 4;
    const int d  = hd & 15;
    const int r  = b * HH + h;

    const int code = idx[r];
    const float ev = e[((size_t)code * 16 + d) * 16 + k];
    const float xv = x[gid];
    out0[gid] = ev;
    out1[gid] = (ev - xv) + xv;
}

// ---------------------------------------------------------------------------
extern "C" void kernel_launch(void* const* d_in, const int* in_sizes, int n_in,
                              void* d_out, int out_size, void* d_ws, size_t ws_size,
                              hipStream_t stream)
{
    (void)in_sizes; (void)n_in; (void)out_size; (void)ws_size;

    const float* x  = (const float*)d_in[0];   // (2048, 128, 16)
    const float* cb = (const float*)d_in[1];   // (16384, 16, 16)
    const float* w  = (const float*)d_in[2];   // (5, 16, 16)

    float* ws    = (float*)d_ws;
    float* e     = ws;                                   // 16384*256 floats
    float* kfeat = e + (size_t)KCODES * 256;             // 16384*192
    float* qfeat = kfeat + (size_t)KCODES * KDIM;        // 16384*192
    int*   idx   = (int*)(qfeat + (size_t)BROWS * KDIM); // 16384 ints

    float* out0 = (float*)d_out;
    float* out1 = out0 + (size_t)2048 * 128 * 16;

    build_e_kfeat<<<KCODES / 16, 256, 0, stream>>>(cb, w, e, kfeat);
    build_qfeat  <<<BROWS * 16 / 256, 256, 0, stream>>>(x, qfeat);
    dist_argmax  <<<BROWS / 128, 256, 0, stream>>>(qfeat, kfeat, idx);
    gather_out   <<<(2048 * 128 * 16) / 256, 256, 0, stream>>>(x, e, idx, out0, out1);
}